// blendv3_75900662055683
// MI455X (gfx1250) — compile-verified
//
#include <hip/hip_runtime.h>
#include <cstdint>
#include <cstddef>

#define NEARZERO 1e-6f
#define UH_LEN   15
#define SCAN_BLK 32     // one wave32 per workgroup -> spread scan across many WGPs
#define ROW      28     // 24 weights + 3 forcings + 1 pad ; 112B row, 16B aligned

// ---------------- gfx1250 async global->LDS builtins -------------------------
// Signatures (from hipcc diagnostics, matching LLVM gfx1250 defs):
//   b128: (v4i addrspace(1)*, v4i addrspace(3)*, imm offset, imm cpol)
//   b32 : (int addrspace(1)*, int addrspace(3)*, imm offset, imm cpol)
typedef int int4v __attribute__((vector_size(16)));

__device__ __forceinline__ void async_ld_b128(const void* g, void* l) {
#if __has_builtin(__builtin_amdgcn_global_load_async_to_lds_b128)
    __builtin_amdgcn_global_load_async_to_lds_b128(
        (__attribute__((address_space(1))) int4v*)(g),
        (__attribute__((address_space(3))) int4v*)(l), 0, 0);
#else
    unsigned lo = (unsigned)(uintptr_t)(__attribute__((address_space(3))) char*)(l);
    asm volatile("global_load_async_to_lds_b128 %0, %1, off" :: "v"(lo), "v"(g) : "memory");
#endif
}
__device__ __forceinline__ void async_ld_b32(const void* g, void* l) {
#if __has_builtin(__builtin_amdgcn_global_load_async_to_lds_b32)
    __builtin_amdgcn_global_load_async_to_lds_b32(
        (__attribute__((address_space(1))) int*)(g),
        (__attribute__((address_space(3))) int*)(l), 0, 0);
#else
    unsigned lo = (unsigned)(uintptr_t)(__attribute__((address_space(3))) char*)(l);
    asm volatile("global_load_async_to_lds_b32 %0, %1, off" :: "v"(lo), "v"(g) : "memory");
#endif
}
#if __has_builtin(__builtin_amdgcn_s_wait_asynccnt)
#define WAIT_ASYNC(n) __builtin_amdgcn_s_wait_asynccnt(n)
#else
#define WAIT_ASYNC(n) asm volatile("s_wait_asynccnt " #n ::: "memory")
#endif

// ---------------- parameter bounds ------------------------------------------
__constant__ float c_lo[32] = {
    0.f, 0.5f, 0.001f, 0.3f, 20.f, -8.f, 1.f, 0.1f, 0.1f, 0.0001f,
    1.f, 0.0001f, 0.1f, 0.1f, 50.f, -8.f, 1.f, -5.f, 0.f, 1.5f,
    0.f, 0.01f, -1.f, 0.f, 0.f, 0.f, 50.f, 50.f, 0.3f, 0.01f, 0.5f, 0.15f};
__constant__ float c_hi[32] = {
    1.f, 3.f, 3.f, 1.f, 300.f, -2.f, 5.f, 200.f, 0.5f, 0.9999f,
    50.f, 0.9999f, 50.f, 100.f, 500.f, -2.f, 5.f, 2.f, 5.f, 3.f,
    5.f, 0.2f, 1.f, 0.4f, 1.f, 1.f, 500.f, 500.f, 20.f, 5.f, 13.f, 1.5f};

// ---------------- kernel 1: per-cell params + unit hydrographs ---------------
__device__ __forceinline__ void make_uh(float alpha, float beta,
                                        float* __restrict__ uh, int g, int G) {
    float lw[UH_LEN];
    float m = -1e30f;
    float rb = __frcp_rn(beta);
#pragma unroll
    for (int k = 0; k < UH_LEN; ++k) {
        float kk = (float)(k + 1);
        lw[k] = (alpha - 1.f) * __logf(kk) - kk * rb;
        m = fmaxf(m, lw[k]);
    }
    float s = 0.f;
#pragma unroll
    for (int k = 0; k < UH_LEN; ++k) { lw[k] = __expf(lw[k] - m); s += lw[k]; }
    float inv = __frcp_rn(s);
#pragma unroll
    for (int k = 0; k < UH_LEN; ++k) uh[(size_t)k * G + g] = lw[k] * inv;
}

__global__ void __launch_bounds__(256)
param_kernel(const float* __restrict__ hp, float* __restrict__ dp,
             float* __restrict__ uh1, float* __restrict__ uh2, int G) {
    int g = blockIdx.x * blockDim.x + threadIdx.x;
    if (g >= G) return;
    float p[32];
#pragma unroll
    for (int i = 0; i < 32; ++i) {
        float x = hp[(size_t)g * 32 + i];
        float s = __frcp_rn(1.f + __expf(-x));
        p[i] = fmaf(s, c_hi[i] - c_lo[i], c_lo[i]);
    }
    const float LN10 = 2.3025850929940457f;
    dp[ 0u*G+g] = p[0];                       // inf_pc
    dp[ 1u*G+g] = p[1];                       // inf_hbv_beta
    dp[ 2u*G+g] = p[2];                       // inf_vic_bexp
    dp[ 3u*G+g] = p[3];                       // inf_hmets_alpha
    dp[ 4u*G+g] = __frcp_rn(p[4]);            // 1/bf1_gr4j_x3
    dp[ 5u*G+g] = __expf(p[5] * LN10);        // 10^bf1_bfc
    dp[ 6u*G+g] = p[6];                       // bf1_bfn
    dp[ 7u*G+g] = p[7];                       // bf1_bfmax
    dp[ 8u*G+g] = p[8];                       // bf1_lam
    dp[ 9u*G+g] = __frcp_rn(expm1f(p[8]));    // 1/expm1(lam)
    dp[10u*G+g] = p[9];                       // bf1_thresh
    dp[11u*G+g] = __frcp_rn(1.f - p[9]);      // 1/(1-thresh)
    dp[12u*G+g] = p[10];                      // max_perc
    dp[13u*G+g] = p[11];                      // perc_sfc
    dp[14u*G+g] = __frcp_rn(1.f - p[11]);     // 1/(1-perc_sfc)
    dp[15u*G+g] = p[12];                      // crise_hbv
    dp[16u*G+g] = p[13];                      // bf2_bfmax
    dp[17u*G+g] = __frcp_rn(p[14]);           // 1/bf2_gr4j_x3
    dp[18u*G+g] = __expf(p[15] * LN10);       // 10^bf2_bfc
    dp[19u*G+g] = p[16];                      // bf2_bfn
    dp[20u*G+g] = p[17];                      // Tbf
    dp[21u*G+g] = p[18];                      // Kf
    dp[22u*G+g] = p[19];                      // ddf_min
    dp[23u*G+g] = p[20];                      // ddf_plus
    dp[24u*G+g] = p[21];                      // Kcum
    dp[25u*G+g] = p[22];                      // Tbm
    dp[26u*G+g] = p[23];                      // swi
    dp[27u*G+g] = p[24] * (1.f - p[25]);      // forest_cover*(1-sparseness)
    dp[28u*G+g] = p[26];                      // max_soilwater1
    dp[29u*G+g] = __frcp_rn(p[26]);
    dp[30u*G+g] = p[27];                      // max_soilwater2
    dp[31u*G+g] = __frcp_rn(p[27]);
    make_uh(p[28], p[29], uh1, g, G);
    make_uh(p[30], p[31], uh2, g, G);
}

// ---------------- softmax-weighted blend of 6 options ------------------------
__device__ __forceinline__ float blend6(const float* __restrict__ r,
                                        float o0, float o1, float o2,
                                        float o3, float o4, float o5) {
    float w0 = r[0], w1 = r[1], w2 = r[2], w3 = r[3], w4 = r[4], w5 = r[5];
    float m = fmaxf(fmaxf(fmaxf(w0, w1), fmaxf(w2, w3)), fmaxf(w4, w5));
    float e0 = __expf(w0 - m), e1 = __expf(w1 - m), e2 = __expf(w2 - m);
    float e3 = __expf(w3 - m), e4 = __expf(w4 - m), e5 = __expf(w5 - m);
    float s = ((e0 + e1) + (e2 + e3)) + (e4 + e5);
    float acc = fmaf(e0, o0, fmaf(e1, o1, fmaf(e2, o2,
                fmaf(e3, o3, fmaf(e4, o4, e5 * o5)))));
    return acc * __frcp_rn(s);
}

// ---------------- kernel 2: sequential hydrology scan ------------------------
// One lane = one grid cell. Forcing+weights for step t+1 are pulled into LDS
// with gfx1250 async global->LDS DMA while step t computes (ASYNCcnt pipeline).
__device__ __forceinline__ void stage_step(const float* __restrict__ wts,
                                           const float* __restrict__ xph,
                                           size_t cell, float* lrow) {
    const char* gw = (const char*)(wts + cell * 24);   // 96B, 16B-aligned
    const char* gx = (const char*)(xph + cell * 3);    // 12B, 4B-aligned
    char* lr = (char*)lrow;
#pragma unroll
    for (int j = 0; j < 6; ++j) async_ld_b128(gw + j * 16, lr + j * 16);
#pragma unroll
    for (int k = 0; k < 3; ++k) async_ld_b32(gx + k * 4, lr + 96 + k * 4);
}

__global__ void __launch_bounds__(SCAN_BLK)
scan_kernel(const float* __restrict__ x_phy, const float* __restrict__ wts,
            const float* __restrict__ dp,
            float* __restrict__ surf_o, float* __restrict__ base_o,
            int T, int G) {
    __shared__ float stage[2][SCAN_BLK][ROW];
    const int tid = threadIdx.x;
    int g = blockIdx.x * SCAN_BLK + tid;
    if (g >= G) g = G - 1;   // duplicate last cell: same inputs -> same writes

    float P[32];
#pragma unroll
    for (int i = 0; i < 32; ++i) P[i] = dp[(size_t)i * G + g];
    const float inf_pc = P[0], hbv_beta = P[1], vic_bexp = P[2], hmets_a = P[3];
    const float x3i1 = P[4], p10c1 = P[5], bfn1 = P[6], bfmx1 = P[7];
    const float lam1 = P[8], em1i = P[9], th1 = P[10], th1i = P[11];
    const float maxperc = P[12], sfc = P[13], sfci = P[14], crise = P[15];
    const float bfmx2 = P[16], x3i2 = P[17], p10c2 = P[18], bfn2 = P[19];
    const float Tbf = P[20], Kf = P[21], ddfmin = P[22], ddfpls = P[23];
    const float Kcum = P[24], Tbm = P[25], swi = P[26], cfc = P[27];
    const float msw1 = P[28], msw1i = P[29], msw2i = P[31];
    (void)P[30];

    float snow = NEARZERO, liq = NEARZERO, cum = NEARZERO;
    float sw1 = NEARZERO, sw2 = NEARZERO;

    stage_step(wts, x_phy, (size_t)0 * G + g, &stage[0][tid][0]);

    for (int t = 0; t < T; ++t) {
        const int b = t & 1;
        if (t + 1 < T) {
            stage_step(wts, x_phy, (size_t)(t + 1) * G + g, &stage[1 - b][tid][0]);
            WAIT_ASYNC(9);   // step t's 9 transfers done; t+1's may be in flight
        } else {
            WAIT_ASYNC(0);
        }
        const float* row = &stage[b][tid][0];
        const float prcp = row[24], temp = row[25], pet = row[26];

        // snow / canopy / melt
        float rain  = (temp >= 0.f) ? prcp : 0.f;
        float snowf = (temp <  0.f) ? prcp : 0.f;
        const float canopy = pet * cfc;
        const float g2 = (row[20] > 0.f) ? 1.f : 0.f;   // round(sigmoid) == sign
        const float g3 = (row[21] > 0.f) ? 1.f : 0.f;
        rain  = fmaxf(rain  - canopy * g2, 0.f);
        snowf = fmaxf(snowf - canopy * g3, 0.f);
        const float refreeze = fminf(liq, Kf * fmaxf(Tbf - temp, 0.f));
        snow += snowf + refreeze;
        liq  -= refreeze;
        const float ddf  = fmaf(ddfpls, 1.f - __expf(-Kcum * cum), ddfmin);
        const float melt = fminf(snow, ddf * fmaxf(temp - Tbm, 0.f));
        snow -= melt;
        cum = (snow < NEARZERO) ? NEARZERO : cum + melt;
        liq += melt;
        const float overflow = fmaxf(liq - swi * snow, 0.f);
        liq -= overflow;
        const float wavail = rain + overflow;

        // infiltration blend
        float sat1 = fminf(fmaxf(sw1 * msw1i, NEARZERO), 1.f);
        float dry1 = fminf(fmaxf(1.f - sat1, NEARZERO), 1.f);
        const float infil = blend6(row + 0,
            wavail * inf_pc,
            wavail * (1.f - __powf(sat1, hbv_beta)),
            wavail * __powf(dry1, vic_bexp),
            wavail * hmets_a * dry1,
            wavail * (1.f - sat1 * sat1),
            wavail);
        sw1 += infil;
        const float excess = fmaxf(sw1 - msw1, 0.f);
        sw1 -= excess;
        const float surfv = wavail - infil + excess;

        // percolation, capillary rise, ET
        sat1 = fminf(fmaxf(sw1 * msw1i, NEARZERO), 1.f);
        const float g0 = (row[18] > 0.f) ? 1.f : 0.f;
        float perc = g0 * maxperc * fmaxf(sat1 - sfc, 0.f) * sfci;
        perc = fminf(perc, sw1);
        sw1 -= perc; sw2 += perc;
        float sat2 = fminf(fmaxf(sw2 * msw2i, NEARZERO), 1.f);
        sat1 = fminf(fmaxf(sw1 * msw1i, NEARZERO), 1.f);
        const float g1 = (row[19] > 0.f) ? 1.f : 0.f;
        const float capi = fminf(g1 * crise * (1.f - sat1) * sat2, sw2);
        sw2 -= capi; sw1 += capi;
        sat1 = fminf(fmaxf(sw1 * msw1i, NEARZERO), 1.f);
        const float et = fminf(pet * sat1, sw1);
        sw1 -= et;
        sat1 = fminf(fmaxf(sw1 * msw1i, NEARZERO), 1.f);

        // baseflow 1 blend
        float q1 = sw1 * x3i1; q1 *= q1; q1 *= q1;               // (sw1/x3)^4
        const float bf1 = fminf(blend6(row + 6,
            sw1 * (1.f - __powf(1.f + q1, -0.25f)),
            p10c1 * sw1,
            bfmx1 * __powf(sat1, bfn1),
            bfmx1 * sat1,
            bfmx1 * expm1f(lam1 * sat1) * em1i,
            bfmx1 * fmaxf(sat1 - th1, 0.f) * th1i), sw1);
        sw1 -= bf1;

        // baseflow 2 blend
        sat2 = fminf(fmaxf(sw2 * msw2i, NEARZERO), 1.f);
        float q2 = sw2 * x3i2; q2 *= q2; q2 *= q2;
        const float bf2 = fminf(blend6(row + 12,
            sw2 * (1.f - __powf(1.f + q2, -0.25f)),
            p10c2 * sw2,
            bfmx2 * __powf(sat2, bfn2),
            bfmx2 * sat2,
            bfmx2 * sat2 * sat2,
            p10c2 * sw2 * sat2), sw2);
        sw2 -= bf2;

        const size_t oo = (size_t)t * G + g;
        surf_o[oo] = surfv;
        base_o[oo] = bf1 + bf2;
    }
}

// ---------------- kernel 3: gated unit-hydrograph routing --------------------
__global__ void __launch_bounds__(256)
route_kernel(const float* __restrict__ surf, const float* __restrict__ base,
             const float* __restrict__ uh1, const float* __restrict__ uh2,
             const float* __restrict__ wts, float* __restrict__ out,
             int T, int G) {
    const int g = blockIdx.x * blockDim.x + threadIdx.x;
    const int t = blockIdx.y;
    if (g >= G) return;
    float qs = 0.f, qb = 0.f;
#pragma unroll
    for (int lag = 0; lag < UH_LEN; ++lag) {
        const int ts = t - lag;
        if (ts >= 0) {
            const size_t o = (size_t)ts * G + g;
            qs = fmaf(uh1[(size_t)lag * G + g], surf[o], qs);
            qb = fmaf(uh2[(size_t)lag * G + g], base[o], qb);
        }
    }
    const size_t o = (size_t)t * G + g;
    const float sv = surf[o], bv = base[o];
    const float* wrow = wts + o * 24;
    const bool g4 = wrow[22] > 0.f;   // gates[:,:,4]  (== round(sigmoid))
    const bool g5 = wrow[23] > 0.f;   // gates[:,:,5]
    out[o] = (g4 ? qs : sv) + (g5 ? qb : bv);
}

// ---------------- launch ----------------------------------------------------
extern "C" void kernel_launch(void* const* d_in, const int* in_sizes, int n_in,
                              void* d_out, int out_size, void* d_ws, size_t ws_size,
                              hipStream_t stream) {
    const float* x_phy = (const float*)d_in[0];   // (T,G,3)
    const float* wts   = (const float*)d_in[1];   // (T,G,24)
    const float* hp    = (const float*)d_in[2];   // (G,32)
    const int G = in_sizes[2] / 32;
    const int T = in_sizes[0] / (3 * G);

    float* dp   = (float*)d_ws;                       // 32*G
    float* uh1  = dp  + (size_t)32 * G;               // 15*G
    float* uh2  = uh1 + (size_t)UH_LEN * G;           // 15*G
    float* surf = uh2 + (size_t)UH_LEN * G;           // T*G
    float* base = surf + (size_t)T * G;               // T*G
    float* out  = (float*)d_out;                      // (T,G)

    const int pgrid = (G + 255) / 256;
    param_kernel<<<dim3(pgrid), dim3(256), 0, stream>>>(hp, dp, uh1, uh2, G);

    const int sgrid = (G + SCAN_BLK - 1) / SCAN_BLK;
    scan_kernel<<<dim3(sgrid), dim3(SCAN_BLK), 0, stream>>>(x_phy, wts, dp,
                                                            surf, base, T, G);

    route_kernel<<<dim3(pgrid, T), dim3(256), 0, stream>>>(surf, base, uh1, uh2,
                                                           wts, out, T, G);
}